// GCNLayer_63608465654232
// MI455X (gfx1250) — compile-verified
//
#include <hip/hip_runtime.h>
#include <cstdint>
#include <cstddef>

#define N_NODES 100000
#define N_EDGES 1250000
#define D_FEAT  64

typedef __attribute__((ext_vector_type(4))) float    v4f;
typedef __attribute__((ext_vector_type(4))) unsigned v4u;

// ---- monotonic float<->uint key: unsigned order == IEEE754 order ------------
__device__ __forceinline__ unsigned f2key(float x) {
    int b = __float_as_int(x);
    return (unsigned)(b ^ ((b >> 31) | 0x80000000));
}
__device__ __forceinline__ float key2f(unsigned k) {
    int b = (k & 0x80000000u) ? (int)(k ^ 0x80000000u) : (int)(~k);
    return __int_as_float(b);
}

// ---- kernel 1: zero the key accumulator and indegree ------------------------
// key 0 is the minimum possible key, so it acts as -inf sentinel.
__global__ __launch_bounds__(256)
void gcn_init_kernel(v4u* __restrict__ agg4, v4u* __restrict__ indeg4) {
    const int ND4 = (N_NODES * D_FEAT) / 4;   // 1,600,000
    const int NI4 = N_NODES / 4;              // 25,000
    int i = blockIdx.x * 256 + threadIdx.x;
    v4u z = 0;
    if (i < ND4) agg4[i] = z;
    if (i < NI4) indeg4[i] = z;
}

// ---- kernel 2: edge scatter-max ---------------------------------------------
// 16 lanes per edge; each lane handles 4 contiguous features (float4 gather,
// fully coalesced 256B row read), then 4 fire-and-forget global_atomic_max_u32.
// Edge-list reads are nontemporal (read-once, keep L2 for inputs+agg).
__global__ __launch_bounds__(256)
void gcn_edge_kernel(const float* __restrict__ inputs,
                     const int*   __restrict__ src,
                     const int*   __restrict__ dst,
                     unsigned*    __restrict__ agg,
                     int*         __restrict__ indeg) {
    int t    = blockIdx.x * 256 + threadIdx.x;
    int e    = t >> 4;
    int lane = t & 15;
    if (e >= N_EDGES) return;

    // software prefetch of a future gathered row (lowers to global_prefetch_b8);
    // 2 prefetching lanes (offsets 0B and 128B) cover the 256B row.
    int ea = e + 2048;
    if (ea < N_EDGES && (lane & 7) == 0) {
        int sa = __builtin_nontemporal_load(&src[ea]);
        __builtin_prefetch(&inputs[(size_t)sa * D_FEAT + lane * 4], 0, 3);
    }

    int s = __builtin_nontemporal_load(&src[e]);
    int d = __builtin_nontemporal_load(&dst[e]);

    const v4f* row = (const v4f*)(inputs + (size_t)s * D_FEAT);
    v4f v = row[lane];

    unsigned* a = agg + (size_t)d * D_FEAT + lane * 4;
    atomicMax(a + 0, f2key(v.x));   // non-returning -> GLOBAL_ATOMIC_MAX_U32
    atomicMax(a + 1, f2key(v.y));
    atomicMax(a + 2, f2key(v.z));
    atomicMax(a + 3, f2key(v.w));

    if (lane == 0) atomicAdd(&indeg[d], 1);
}

// ---- kernel 3: finalize + concat --------------------------------------------
// thread = (node, 4-feature chunk). agg/indeg are dead after this pass -> NT
// loads; output is write-once streaming (51 MB) -> NT stores so it doesn't
// evict the L2-resident working set.
__global__ __launch_bounds__(256)
void gcn_final_kernel(const float*    __restrict__ inputs,
                      const unsigned* __restrict__ agg,
                      const int*      __restrict__ indeg,
                      float*          __restrict__ out) {
    int t = blockIdx.x * 256 + threadIdx.x;
    if (t >= N_NODES * 16) return;
    int n = t >> 4;
    int c = t & 15;

    v4f vin = ((const v4f*)(inputs + (size_t)n * D_FEAT))[c];
    v4u k   = __builtin_nontemporal_load(((const v4u*)(agg + (size_t)n * D_FEAT)) + c);

    v4f vagg;
    vagg.x = key2f(k.x);
    vagg.y = key2f(k.y);
    vagg.z = key2f(k.z);
    vagg.w = key2f(k.w);

    int deg = __builtin_nontemporal_load(&indeg[n]);
    v4f vsel = (deg > 0) ? vagg : vin;

    float* o = out + (size_t)n * (2 * D_FEAT);
    __builtin_nontemporal_store(vin,  ((v4f*)o) + c);
    __builtin_nontemporal_store(vsel, ((v4f*)(o + D_FEAT)) + c);
}

extern "C" void kernel_launch(void* const* d_in, const int* in_sizes, int n_in,
                              void* d_out, int out_size, void* d_ws, size_t ws_size,
                              hipStream_t stream) {
    const float* inputs = (const float*)d_in[0];
    const int*   src    = (const int*)d_in[1];
    const int*   dst    = (const int*)d_in[2];
    float*       out    = (float*)d_out;

    // workspace layout: [ N*D u32 keys | N i32 indegree ]
    unsigned* agg   = (unsigned*)d_ws;
    int*      indeg = (int*)((char*)d_ws + (size_t)N_NODES * D_FEAT * sizeof(unsigned));

    // 1) init accumulator (deterministic: re-done every call)
    {
        int total  = (N_NODES * D_FEAT) / 4;   // covers indeg range too (25k < 1.6M)
        int blocks = (total + 255) / 256;
        gcn_init_kernel<<<blocks, 256, 0, stream>>>((v4u*)agg, (v4u*)indeg);
    }
    // 2) edge scatter-max: 16 threads per edge
    {
        long long threads = (long long)N_EDGES * 16;
        int blocks = (int)((threads + 255) / 256);
        gcn_edge_kernel<<<blocks, 256, 0, stream>>>(inputs, src, dst, agg, indeg);
    }
    // 3) finalize + concat
    {
        long long threads = (long long)N_NODES * 16;
        int blocks = (int)((threads + 255) / 256);
        gcn_final_kernel<<<blocks, 256, 0, stream>>>(inputs, agg, indeg, out);
    }
}